// SemiSeparableTokenMixing_43748536877649
// MI455X (gfx1250) — compile-verified
//
#include <hip/hip_runtime.h>
#include <stdint.h>

// Problem constants (fixed by the reference).
#define BATCH  4
#define SEQ    4096
#define DIM    1024
#define NCH    32                // chunks along seq
#define CH     (SEQ / NCH)       // 128 timesteps per chunk
#define TT     8                 // timesteps per LDS tile
#define NT     (CH / TT)         // 16 tiles per chunk
#define STAGES 4                 // LDS pipeline depth (3 tiles prefetched ahead)
#define WAVES  4
#define BLK    (WAVES * 32)      // 128 threads / block

// LDS byte offset of a __shared__ object: for the LDS aperture the low 32 bits
// of the flat address are the wave-relative LDS byte address (ISA 10.2).
__device__ __forceinline__ uint32_t lds_off(const void* p) {
  return (uint32_t)(uintptr_t)p;
}

#define WAIT_ASYNC(n) asm volatile("s_wait_asynccnt %0" :: "i"(n) : "memory")

__device__ __forceinline__ void async_b128(uint32_t ldsb, const float* g) {
  // CDNA5 async copy: 16B per lane into LDS, tracked by ASYNCcnt (default RT).
  asm volatile("global_load_async_to_lds_b128 %0, %1, off"
               :: "v"(ldsb), "v"(g) : "memory");
}

__device__ __forceinline__ void async_b128_lu(uint32_t ldsb, const float* g) {
  // Last-use temporal hint: data is consumed exactly once (pass 3).
  asm volatile("global_load_async_to_lds_b128 %0, %1, off th:TH_LOAD_LU"
               :: "v"(ldsb), "v"(g) : "memory");
}

// Copy an 8(row=t) x 32(lane=d) float tile (global row stride = DIM floats)
// into LDS row-major [t][lane]. 2 instructions, 32 lanes x 16B each.
template <bool LASTUSE>
__device__ __forceinline__ void async_tile(const float* gbase, uint32_t ldsb, int lane) {
  const int row = lane >> 3;          // 0..3
  const int col = (lane & 7) << 2;    // float column 0,4,...,28
  if (LASTUSE) {
    async_b128_lu(ldsb + (uint32_t)(((row)     * 32 + col) << 2), gbase + (size_t)(row)     * DIM + col);
    async_b128_lu(ldsb + (uint32_t)(((row + 4) * 32 + col) << 2), gbase + (size_t)(row + 4) * DIM + col);
  } else {
    async_b128(ldsb + (uint32_t)(((row)     * 32 + col) << 2), gbase + (size_t)(row)     * DIM + col);
    async_b128(ldsb + (uint32_t)(((row + 4) * 32 + col) << 2), gbase + (size_t)(row + 4) * DIM + col);
  }
}

// Wait until all tiles except `ahead` most-recent are complete.
// OPS = async instructions per tile. Async loads complete in issue order,
// so ASYNCcnt <= ahead*OPS guarantees the tile being consumed has landed.
template <int OPS>
__device__ __forceinline__ void wait_tiles(int ahead) {
  switch (ahead) {
    case 3:  WAIT_ASYNC(3 * OPS); break;
    case 2:  WAIT_ASYNC(2 * OPS); break;
    case 1:  WAIT_ASYNC(1 * OPS); break;
    default: WAIT_ASYNC(0);       break;
  }
}

// ---------------- Pass 1: per-chunk (prod A, h_local with h_in = 0) ----------------
__global__ __launch_bounds__(BLK)
void sssm_pass1(const float* __restrict__ A, const float* __restrict__ Bm,
                float* __restrict__ aprodW, float* __restrict__ hW) {
  __shared__ float sm[WAVES * STAGES * 2 * TT * 32];   // 4*4*2*256 floats = 32 KB
  const int w    = threadIdx.x >> 5;
  const int lane = threadIdx.x & 31;
  const int bc   = blockIdx.x;                 // 8 dimblk * 32 chunk * 4 batch = 1024
  const int d0   = ((bc & 7) << 7) + (w << 5);
  const int chk  = (bc >> 3) & 31;
  const int b    = bc >> 8;
  const int d    = d0 + lane;

  const size_t gchan = ((size_t)b * SEQ + (size_t)chk * CH) * DIM + d0;
  const float* gA = A  + gchan;
  const float* gB = Bm + gchan;

  float* smw = &sm[w * (STAGES * 2 * TT * 32)];
  const uint32_t smwb   = lds_off(smw);
  const uint32_t stageB = 2 * TT * 32 * 4;
  const uint32_t arrB   = TT * 32 * 4;

  // Prologue: prefetch tiles 0..STAGES-2.
#pragma unroll
  for (int p = 0; p < STAGES - 1; ++p) {
    const size_t go = (size_t)p * TT * DIM;
    const uint32_t sb = smwb + (uint32_t)p * stageB;
    async_tile<false>(gA + go, sb + 0 * arrB, lane);
    async_tile<false>(gB + go, sb + 1 * arrB, lane);
  }

  float h = 0.0f, ap = 1.0f;
  for (int tile = 0; tile < NT; ++tile) {
    const int s   = tile & (STAGES - 1);
    const int nxt = tile + (STAGES - 1);
    if (nxt < NT) {
      const uint32_t sb = smwb + (uint32_t)(nxt & (STAGES - 1)) * stageB;
      const size_t  go  = (size_t)nxt * TT * DIM;
      async_tile<false>(gA + go, sb + 0 * arrB, lane);
      async_tile<false>(gB + go, sb + 1 * arrB, lane);
    }
    int ahead = NT - 1 - tile; if (ahead > STAGES - 1) ahead = STAGES - 1;
    wait_tiles<4>(ahead);                       // 4 async ops per tile (2 arrays x 2)
    const float* tA = smw + (s * 2 + 0) * (TT * 32);
    const float* tB = smw + (s * 2 + 1) * (TT * 32);
#pragma unroll
    for (int t = 0; t < TT; ++t) {
      const float a  = tA[t * 32 + lane];
      const float bb = tB[t * 32 + lane];
      ap *= a;
      h = fmaf(a, h, bb);
    }
  }
  const size_t ci = ((size_t)b * NCH + chk) * DIM + d;
  aprodW[ci] = ap;
  hW[ci]     = h;
}

// ---------------- Pass 2: serial combine of 32 chunk carries per channel ----------------
__global__ void sssm_pass2(const float* __restrict__ aprodW, const float* __restrict__ hW,
                           float* __restrict__ hinW) {
  const int tid = blockIdx.x * blockDim.x + threadIdx.x;   // b*DIM + d
  if (tid >= BATCH * DIM) return;
  const int b = tid >> 10;
  const int d = tid & (DIM - 1);
  float h = 0.0f;
  for (int c = 0; c < NCH; ++c) {
    const size_t ci = ((size_t)b * NCH + c) * DIM + d;
    hinW[ci] = h;                               // h entering chunk c
    h = fmaf(aprodW[ci], h, hW[ci]);            // h leaving chunk c
  }
}

// ---------------- Pass 3: re-scan chunk with correct h_in, emit y = C*h ----------------
__global__ __launch_bounds__(BLK)
void sssm_pass3(const float* __restrict__ A, const float* __restrict__ Bm,
                const float* __restrict__ Cm, const float* __restrict__ hinW,
                float* __restrict__ out) {
  __shared__ float sm[WAVES * STAGES * 3 * TT * 32];   // 4*4*3*256 floats = 48 KB
  const int w    = threadIdx.x >> 5;
  const int lane = threadIdx.x & 31;
  const int bc   = blockIdx.x;
  const int d0   = ((bc & 7) << 7) + (w << 5);
  const int chk  = (bc >> 3) & 31;
  const int b    = bc >> 8;
  const int d    = d0 + lane;

  const size_t gchan = ((size_t)b * SEQ + (size_t)chk * CH) * DIM + d0;
  const float* gA = A  + gchan;
  const float* gB = Bm + gchan;
  const float* gC = Cm + gchan;
  float*       gO = out + gchan;

  float* smw = &sm[w * (STAGES * 3 * TT * 32)];
  const uint32_t smwb   = lds_off(smw);
  const uint32_t stageB = 3 * TT * 32 * 4;
  const uint32_t arrB   = TT * 32 * 4;

#pragma unroll
  for (int p = 0; p < STAGES - 1; ++p) {
    const size_t go = (size_t)p * TT * DIM;
    const uint32_t sb = smwb + (uint32_t)p * stageB;
    async_tile<true>(gA + go, sb + 0 * arrB, lane);
    async_tile<true>(gB + go, sb + 1 * arrB, lane);
    async_tile<true>(gC + go, sb + 2 * arrB, lane);
  }

  float h = hinW[((size_t)b * NCH + chk) * DIM + d];

  for (int tile = 0; tile < NT; ++tile) {
    const int s   = tile & (STAGES - 1);
    const int nxt = tile + (STAGES - 1);
    if (nxt < NT) {
      const uint32_t sb = smwb + (uint32_t)(nxt & (STAGES - 1)) * stageB;
      const size_t  go  = (size_t)nxt * TT * DIM;
      async_tile<true>(gA + go, sb + 0 * arrB, lane);
      async_tile<true>(gB + go, sb + 1 * arrB, lane);
      async_tile<true>(gC + go, sb + 2 * arrB, lane);
    }
    int ahead = NT - 1 - tile; if (ahead > STAGES - 1) ahead = STAGES - 1;
    wait_tiles<6>(ahead);                       // 6 async ops per tile (3 arrays x 2)
    const float* tA = smw + (s * 3 + 0) * (TT * 32);
    const float* tB = smw + (s * 3 + 1) * (TT * 32);
    const float* tC = smw + (s * 3 + 2) * (TT * 32);
    float* o = gO + (size_t)tile * TT * DIM + lane;
#pragma unroll
    for (int t = 0; t < TT; ++t) {
      const float a  = tA[t * 32 + lane];
      const float bb = tB[t * 32 + lane];
      const float c  = tC[t * 32 + lane];
      h = fmaf(a, h, bb);
      __builtin_nontemporal_store(c * h, o + (size_t)t * DIM);  // y never re-read
    }
  }
}

extern "C" void kernel_launch(void* const* d_in, const int* in_sizes, int n_in,
                              void* d_out, int out_size, void* d_ws, size_t ws_size,
                              hipStream_t stream) {
  // setup_inputs order: x (unused), B, C, A — all float32.
  const float* Bm = (const float*)d_in[1];
  const float* Cm = (const float*)d_in[2];
  const float* A  = (const float*)d_in[3];
  float* out = (float*)d_out;

  // Workspace: 3 carry arrays of BATCH*NCH*DIM floats (1.5 MB total).
  float* aprodW = (float*)d_ws;
  float* hW     = aprodW + (size_t)BATCH * NCH * DIM;
  float* hinW   = hW     + (size_t)BATCH * NCH * DIM;

  const int nblk = BATCH * NCH * (DIM / BLK);   // 1024 blocks, 4096 waves
  sssm_pass1<<<nblk, BLK, 0, stream>>>(A, Bm, aprodW, hW);
  sssm_pass2<<<(BATCH * DIM + 255) / 256, 256, 0, stream>>>(aprodW, hW, hinW);
  sssm_pass3<<<nblk, BLK, 0, stream>>>(A, Bm, Cm, hinW, out);
}